// EdgesAugmentedLayer_56538949484714
// MI455X (gfx1250) — compile-verified
//
#include <hip/hip_runtime.h>
#include <hip/hip_bf16.h>
#include <stdint.h>

// Problem constants (match reference: B=32, M=4096, F=64, N=MAX_NUM_ATOMS=128)
#define NATOMS   128
#define MEDGES   4096     // edges per batch (M)
#define NF       64       // features per edge (F)
#define THREADS  256      // 8 wave32 per block
#define EPB      128      // edges per block -> 128*64*4 = 32 KB LDS staging

typedef int v4i __attribute__((ext_vector_type(4)));
typedef __attribute__((address_space(1))) v4i* gv4i_t;   // global (device) int4*
typedef __attribute__((address_space(3))) v4i* lv4i_t;   // LDS int4*

// ---- CDNA5 async global->LDS copy (ASYNCcnt path) -------------------------
__device__ __forceinline__ void async_ld_b128(const char* g, char* l) {
#if __has_builtin(__builtin_amdgcn_global_load_async_to_lds_b128)
  __builtin_amdgcn_global_load_async_to_lds_b128(
      (gv4i_t)(void*)g, (lv4i_t)(void*)l, 0, 0);
#else
  unsigned loff = (unsigned)(size_t)l;   // low 32 bits of generic LDS addr = LDS offset
  asm volatile("global_load_async_to_lds_b128 %0, %1, off"
               :: "v"(loff), "v"(g) : "memory");
#endif
}

__device__ __forceinline__ void wait_async0() {
#if __has_builtin(__builtin_amdgcn_s_wait_asynccnt)
  __builtin_amdgcn_s_wait_asynccnt(0);
#else
  asm volatile("s_wait_asynccnt 0x0" ::: "memory");
#endif
}

// ---- Device-scope, no-return FP32 global atomic add (STOREcnt path) -------
// Guarantees native global_atomic_add_f32 (L2 atomic units; 128 MB output is
// L2-resident on MI455X's 192 MB L2) instead of a CAS loop.
__device__ __forceinline__ void gatomic_add_f32(float* base, unsigned byteoff, float v) {
  asm volatile("global_atomic_add_f32 %0, %1, %2 scope:SCOPE_DEV"
               :: "v"(byteoff), "v"(v), "s"(base) : "memory");
}

// ---- Kernel 1: zero-init the 128 MB output with b128 stores ---------------
__global__ __launch_bounds__(THREADS)
void zero_kernel(float4* __restrict__ out, int n4) {
  int i = blockIdx.x * blockDim.x + threadIdx.x;
  if (i < n4) out[i] = make_float4(0.f, 0.f, 0.f, 0.f);
}

// ---- Kernel 2: scatter-add edges into [B, N, N, F] ------------------------
__global__ __launch_bounds__(THREADS)
void scatter_kernel(const float* __restrict__ feats,
                    const long long* __restrict__ pairs,
                    float* __restrict__ out, int edges) {
  __shared__ float    sfeat[EPB * NF];   // 32 KB edge-feature staging
  __shared__ unsigned soff[EPB];         // per-edge output byte offset

  const int tid = threadIdx.x;
  const int e0  = blockIdx.x * EPB;

  // Phase A1: kick off async copy of this block's edge features (2048 x b128).
  const char* src = (const char*)(feats + (size_t)e0 * NF);
  char*       dst = (char*)sfeat;
  const int nEdges    = (edges - e0) < EPB ? (edges - e0) : EPB;
  const int featBytes = nEdges * NF * 4;
  #pragma unroll
  for (int k = 0; k < (EPB * NF * 4) / (16 * THREADS); ++k) {  // 8 iterations
    int off = (tid + k * THREADS) * 16;
    if (off < featBytes) async_ld_b128(src + off, dst + off);
  }

  // Phase A2 (overlapped with async copy): compute output byte offsets.
  if (tid < EPB && (e0 + tid) < edges) {
    int ge = e0 + tid;
    unsigned b = (unsigned)ge / MEDGES;
    unsigned i = (unsigned)pairs[(size_t)ge * 2 + 0];
    unsigned j = (unsigned)pairs[(size_t)ge * 2 + 1];
    // max offset = 32*128*128*64*4 = 2^27 bytes -> fits u32
    soff[tid] = ((b * NATOMS + i) * NATOMS + j) * (NF * 4u);
  }

  wait_async0();       // own async LDS writes landed
  __syncthreads();     // visible to whole workgroup

  // Phase B: 8 waves x 16 edges; lane l handles floats [2l, 2l+1] of the edge.
  const int wave = tid >> 5, lane = tid & 31;
  #pragma unroll
  for (int e = 0; e < EPB / (THREADS / 32); ++e) {   // 16 per wave
    int edge = wave * (EPB / (THREADS / 32)) + e;
    if (e0 + edge >= edges) break;
    unsigned ob = soff[edge] + (unsigned)(lane * 8); // coalesced 256B per wave
    float2 v = *(const float2*)(&sfeat[edge * NF + lane * 2]); // ds_load_b64, conflict-free
    gatomic_add_f32(out, ob,     v.x);
    gatomic_add_f32(out, ob + 4, v.y);
  }
}

extern "C" void kernel_launch(void* const* d_in, const int* in_sizes, int n_in,
                              void* d_out, int out_size, void* d_ws, size_t ws_size,
                              hipStream_t stream) {
  const float*     feats = (const float*)d_in[0];       // [B, M, F] f32
  const long long* pairs = (const long long*)d_in[1];   // [B, M, 2] i64
  float*           out   = (float*)d_out;               // [B, N, N, F] f32

  int edges = in_sizes[1] / 2;        // B*M = 131072
  int n4    = out_size / 4;           // 8,388,608 float4

  zero_kernel<<<(n4 + THREADS - 1) / THREADS, THREADS, 0, stream>>>((float4*)out, n4);

  int blocks = (edges + EPB - 1) / EPB;   // 1024
  scatter_kernel<<<blocks, THREADS, 0, stream>>>(feats, pairs, out, edges);
}